// GraphWholeMessage_81698867904550
// MI455X (gfx1250) — compile-verified
//
#include <hip/hip_runtime.h>
#include <math.h>

#define BS    16
#define TT    20
#define NND   512
#define EMB   64
#define LAG   8
#define STEPS 12
#define TR    19
#define K2    2

typedef float v2f __attribute__((ext_vector_type(2)));
typedef float v8f __attribute__((ext_vector_type(8)));

__device__ __forceinline__ v8f wmma4(v2f a, v2f b, v8f c) {
  // D = A(16x4, f32) x B(4x16, f32) + C(16x16, f32)
  return __builtin_amdgcn_wmma_f32_16x16x4_f32(false, a, false, b, (short)0, c, false, false);
}

// ---------------------------------------------------------------------------
// crow[k,b,t,n] = adj[k,b,s,n] + (1/N) * sum_m adj[k,b,s,m]*adj[k,b,m,n],
// with s = skill_seq[b,t]; t = 0..18. Folds the s-stack sum of the einsum.
// ---------------------------------------------------------------------------
__global__ void k_crow(const float* __restrict__ adj, const int* __restrict__ skill,
                       float* __restrict__ crow) {
  int t = blockIdx.x, b = blockIdx.y, k = blockIdx.z;
  int s = skill[b * TT + t];
  const float* A = adj + ((size_t)(k * BS + b)) * NND * NND;
  __shared__ float arow[NND];
  for (int i = threadIdx.x; i < NND; i += blockDim.x) arow[i] = A[(size_t)s * NND + i];
  __syncthreads();
  for (int n = threadIdx.x; n < NND; n += blockDim.x) {
    float acc = 0.f;
    for (int m = 0; m < NND; ++m) acc = fmaf(arow[m], A[(size_t)m * NND + n], acc);
    crow[(((size_t)k * BS + b) * TR + t) * NND + n] = arow[n] + acc * (1.0f / NND);
  }
}

// ---------------------------------------------------------------------------
// hidden0 scatter: hidden[b, s0[b,l], :] = emb[s0]*lp   (only last occurrence)
// ---------------------------------------------------------------------------
__global__ void k_init_hidden(const int* __restrict__ skill, const int* __restrict__ label,
                              const float* __restrict__ emb, float* __restrict__ hidden) {
  int b = blockIdx.x;
  float* H = hidden + (size_t)b * NND * EMB;
  for (int i = threadIdx.x; i < NND * EMB; i += blockDim.x) H[i] = 0.f;
  __syncthreads();
  for (int idx = threadIdx.x; idx < LAG * EMB; idx += blockDim.x) {
    int l = idx / EMB, e = idx % EMB;
    int s = skill[b * TT + l];
    bool last = true;
    for (int j = l + 1; j < LAG; ++j)
      if (skill[b * TT + j] == s) { last = false; break; }
    if (!last) continue;
    int lab = label[b * TT + l];
    float lp = (lab == 0) ? -1.f : ((lab == -1) ? 0.f : (float)lab);
    H[s * EMB + e] = emb[s * EMB + e] * lp;
  }
}

// ---------------------------------------------------------------------------
// u[b,l,e] = b1[e] + sum_i hidden[b, sh, i] * w1[e, i]      (first half of fc1)
// ---------------------------------------------------------------------------
__global__ void k_u(const float* __restrict__ hidden, const int* __restrict__ skill,
                    const float* __restrict__ w1, const float* __restrict__ b1,
                    float* __restrict__ u, int step) {
  int blk = blockIdx.x;               // b*LAG + l
  int b = blk / LAG, l = blk % LAG;
  int e = threadIdx.x;                // 64
  int sh = skill[b * TT + step + l];
  const float* h = hidden + ((size_t)b * NND + sh) * EMB;
  float acc = b1[e];
  for (int i = 0; i < EMB; ++i) acc = fmaf(h[i], w1[e * 2 * EMB + i], acc);
  u[(b * LAG + l) * EMB + e] = acc;
}

// ---------------------------------------------------------------------------
// v = hidden @ W1[:,64:128]^T : GEMM M=8192, N=64, K=64 via v_wmma_f32_16x16x4
// 1 wave = one 16x16 D tile; block of 8 waves covers 32 rows x 64 cols.
// ---------------------------------------------------------------------------
__global__ void __launch_bounds__(256) k_v(const float* __restrict__ hidden,
                                           const float* __restrict__ w1,
                                           float* __restrict__ v) {
  int lane = threadIdx.x & 31;
  int wave = threadIdx.x >> 5;
  int m0 = blockIdx.x * 32 + (wave >> 2) * 16;
  int e0 = (wave & 3) * 16;
  int lr = lane & 15;
  int khalf = (lane < 16) ? 0 : 2;       // A/B frag: lanes 0-15 hold K 0/1, 16-31 hold K 2/3
  const float* arow = hidden + (size_t)(m0 + lr) * EMB;
  const float* bw = w1 + (size_t)(e0 + lr) * (2 * EMB) + EMB;  // second-half cols
  v8f c = {};
  for (int k0 = 0; k0 < EMB; k0 += 4) {
    v2f a, bb;
    a.x  = arow[k0 + khalf];  a.y  = arow[k0 + khalf + 1];
    bb.x = bw[k0 + khalf];    bb.y = bw[k0 + khalf + 1];
    c = wmma4(a, bb, c);
  }
  int mbase = m0 + ((lane < 16) ? 0 : 8);   // D: VGPR r -> M=r (+8 for hi lanes)
  #pragma unroll
  for (int r = 0; r < 8; ++r)
    v[(size_t)(mbase + r) * EMB + e0 + lr] = c[r];
}

// ---------------------------------------------------------------------------
// Fused per-step message kernel. Block = (b, 16-node tile).
//  m1 = tanh(u[l] + v[n]);  m2 = tanh(m1 @ W2^T + b2) * exp(-dtw[l]);
//  agg[k,n,e] = sum_l crow[k,b,step+l,n] * m2[l,n,e];  hnext = mean_k agg.
// M-tile mt of the 128x64 GEMM is exactly lag l = mt (rows r = l*16 + j).
// ---------------------------------------------------------------------------
__global__ void __launch_bounds__(256) k_step(
    int step,
    const float* __restrict__ ubuf, const float* __restrict__ vbuf,
    const float* __restrict__ crow, const int* __restrict__ timeq,
    const float* __restrict__ w2, const float* __restrict__ b2,
    float* __restrict__ agg, float* __restrict__ hnext) {
  int nt = blockIdx.x;   // 0..31
  int b  = blockIdx.y;   // 0..15
  int tid = threadIdx.x;

  __shared__ float lu[LAG][EMB];
  __shared__ float lv[16][EMB];
  __shared__ float ew[LAG];
  __shared__ float m1[LAG * 16][EMB];
  __shared__ float crw[K2][LAG][16];
  __shared__ float aggL[K2][16][EMB];

  for (int i = tid; i < LAG * EMB; i += 256)
    lu[i / EMB][i % EMB] = ubuf[(b * LAG) * EMB + i];
  for (int i = tid; i < 16 * EMB; i += 256)
    lv[i / EMB][i % EMB] = vbuf[((size_t)b * NND + nt * 16) * EMB + i];
  if (tid < LAG) {
    float t0 = (float)timeq[b * TT + step + tid];
    float t1 = (float)timeq[b * TT + step + LAG];
    float dtw = logf(fabsf(t0 - t1) + 1e-6f) / logf(5.0f);
    ew[tid] = expf(-dtw);
  }
  for (int i = tid; i < K2 * LAG * 16; i += 256) {
    int k = i / (LAG * 16), l = (i / 16) % LAG, j = i % 16;
    crw[k][l][j] = crow[(((size_t)k * BS + b) * TR + (step + l)) * NND + nt * 16 + j];
  }
  for (int i = tid; i < K2 * 16 * EMB; i += 256) ((float*)aggL)[i] = 0.f;
  __syncthreads();

  for (int i = tid; i < LAG * 16 * EMB; i += 256) {
    int r = i / EMB, e = i % EMB;
    m1[r][e] = tanhf(lu[r >> 4][e] + lv[r & 15][e]);
  }
  __syncthreads();

  int lane = tid & 31;
  int mt = tid >> 5;                 // wave id == lag l
  int lr = lane & 15;
  int khalf = (lane < 16) ? 0 : 2;
  float sc = ew[mt];
  for (int nt2 = 0; nt2 < 4; ++nt2) {
    int e0 = nt2 * 16;
    v8f c = {};
    for (int k0 = 0; k0 < EMB; k0 += 4) {
      v2f a, bb;
      a.x  = m1[mt * 16 + lr][k0 + khalf];
      a.y  = m1[mt * 16 + lr][k0 + khalf + 1];
      bb.x = w2[(e0 + lr) * EMB + k0 + khalf];
      bb.y = w2[(e0 + lr) * EMB + k0 + khalf + 1];
      c = wmma4(a, bb, c);
    }
    int jbase = (lane < 16) ? 0 : 8;
    int ec = e0 + lr;
    float bias = b2[ec];
    #pragma unroll
    for (int r = 0; r < 8; ++r) {
      int j = jbase + r;
      float val = tanhf(c[r] + bias) * sc;
      atomicAdd(&aggL[0][j][ec], crw[0][mt][j] * val);   // ds_add_f32
      atomicAdd(&aggL[1][j][ec], crw[1][mt][j] * val);
    }
  }
  __syncthreads();

  for (int i = tid; i < 16 * EMB; i += 256) {
    int j = i / EMB, e = i % EMB;
    int n = nt * 16 + j;
    float a0 = aggL[0][j][e], a1 = aggL[1][j][e];
    agg[(((size_t)b * K2 + 0) * NND + n) * EMB + e] = a0;
    agg[(((size_t)b * K2 + 1) * NND + n) * EMB + e] = a1;
    hnext[((size_t)b * NND + n) * EMB + e] = 0.5f * (a0 + a1);
  }
}

// ---------------------------------------------------------------------------
// Head: only rows n = tgt[b] ever reach the output. 32 rows per step.
// ---------------------------------------------------------------------------
__device__ __forceinline__ float block64_sum(float v, float* red, int e) {
  red[e] = v; __syncthreads();
  for (int st = 32; st > 0; st >>= 1) {
    if (e < st) red[e] += red[e + st];
    __syncthreads();
  }
  float s = red[0]; __syncthreads();
  return s;
}

__global__ void k_head(int step, const float* __restrict__ agg, const int* __restrict__ skill,
                       const float* __restrict__ fw1, const float* __restrict__ fb1,
                       const float* __restrict__ g1,  const float* __restrict__ bb1,
                       const float* __restrict__ fw2, const float* __restrict__ fb2,
                       const float* __restrict__ g2,  const float* __restrict__ bb2,
                       const float* __restrict__ fw3, const float* __restrict__ fb3,
                       float* __restrict__ out) {
  int blk = blockIdx.x;              // b*2 + k
  int b = blk >> 1, k = blk & 1;
  int e = threadIdx.x;               // 64
  int tgt = skill[b * TT + LAG + step];
  __shared__ float sx[EMB], sv[EMB], red[EMB];
  const float* row = agg + (((size_t)b * K2 + k) * NND + tgt) * EMB;

  sx[e] = row[e];
  __syncthreads();
  float acc = fb1[e];
  for (int i = 0; i < EMB; ++i) acc = fmaf(sx[i], fw1[e * EMB + i], acc);
  float h1 = sx[e] + (acc > 0.f ? acc : 0.01f * acc);

  float mean = block64_sum(h1, red, e) * (1.0f / EMB);
  float d = h1 - mean;
  float var = block64_sum(d * d, red, e) * (1.0f / EMB);
  float ln1 = (h1 - mean) * rsqrtf(var + 1e-5f) * g1[e] + bb1[e];
  sv[e] = ln1; __syncthreads();

  float acc2 = fb2[e];
  for (int i = 0; i < EMB; ++i) acc2 = fmaf(sv[i], fw2[e * EMB + i], acc2);
  float h2 = h1 + (acc2 > 0.f ? acc2 : 0.01f * acc2);

  float mean2 = block64_sum(h2, red, e) * (1.0f / EMB);
  float d2 = h2 - mean2;
  float var2 = block64_sum(d2 * d2, red, e) * (1.0f / EMB);
  float ln2 = (h2 - mean2) * rsqrtf(var2 + 1e-5f) * g2[e] + bb2[e];

  float l0 = block64_sum(ln2 * fw3[e],       red, e) + fb3[0];
  float l1 = block64_sum(ln2 * fw3[EMB + e], red, e) + fb3[1];
  if (e == 0) {
    float m = fmaxf(l0, l1);
    float e0 = expf(l0 - m), e1 = expf(l1 - m);
    out[((size_t)step * BS + b) * K2 + k] = e1 / (e0 + e1);
  }
}

extern "C" void kernel_launch(void* const* d_in, const int* in_sizes, int n_in,
                              void* d_out, int out_size, void* d_ws, size_t ws_size,
                              hipStream_t stream) {
  const int*   skill = (const int*)d_in[0];
  const int*   timeq = (const int*)d_in[1];
  const int*   label = (const int*)d_in[2];
  const float* adj   = (const float*)d_in[3];
  const float* emb   = (const float*)d_in[4];
  const float* w1    = (const float*)d_in[5];
  const float* b1    = (const float*)d_in[6];
  const float* w2    = (const float*)d_in[7];
  const float* b2    = (const float*)d_in[8];
  const float* fw1   = (const float*)d_in[9];
  const float* fb1   = (const float*)d_in[10];
  const float* g1    = (const float*)d_in[11];
  const float* bb1   = (const float*)d_in[12];
  const float* fw2   = (const float*)d_in[13];
  const float* fb2   = (const float*)d_in[14];
  const float* g2    = (const float*)d_in[15];
  const float* bb2   = (const float*)d_in[16];
  const float* fw3   = (const float*)d_in[17];
  const float* fb3   = (const float*)d_in[18];
  float* out = (float*)d_out;

  float* ws = (float*)d_ws;
  size_t o = 0;
  float* crow = ws + o;  o += (size_t)K2 * BS * TR * NND;     // 311,296
  float* hA   = ws + o;  o += (size_t)BS * NND * EMB;         // 524,288
  float* hB   = ws + o;  o += (size_t)BS * NND * EMB;
  float* ubuf = ws + o;  o += (size_t)BS * LAG * EMB;
  float* vbuf = ws + o;  o += (size_t)BS * NND * EMB;
  float* agg  = ws + o;  o += (size_t)BS * K2 * NND * EMB;    // total ~11.8 MB

  hipLaunchKernelGGL(k_crow, dim3(TR, BS, K2), dim3(256), 0, stream, adj, skill, crow);
  hipLaunchKernelGGL(k_init_hidden, dim3(BS), dim3(256), 0, stream, skill, label, emb, hA);

  for (int s = 0; s < STEPS; ++s) {
    float* cur = (s & 1) ? hB : hA;
    float* nxt = (s & 1) ? hA : hB;
    hipLaunchKernelGGL(k_u, dim3(BS * LAG), dim3(EMB), 0, stream, cur, skill, w1, b1, ubuf, s);
    hipLaunchKernelGGL(k_v, dim3(BS * NND / 32), dim3(256), 0, stream, cur, w1, vbuf);
    hipLaunchKernelGGL(k_step, dim3(NND / 16, BS), dim3(256), 0, stream,
                       s, ubuf, vbuf, crow, timeq, w2, b2, agg, nxt);
    hipLaunchKernelGGL(k_head, dim3(BS * K2), dim3(EMB), 0, stream,
                       s, agg, skill, fw1, fb1, g1, bb1, fw2, fb2, g2, bb2, fw3, fb3, out);
  }
}